// PointnetFPModule_63144609186371
// MI455X (gfx1250) — compile-verified
//
#include <hip/hip_runtime.h>
#include <hip/hip_bf16.h>
#include <math.h>

typedef __attribute__((ext_vector_type(2))) float v2f;
typedef __attribute__((ext_vector_type(8))) float v8f;

constexpr int Bb = 16, Nn = 4096, Mm = 1024, C1 = 256, C2 = 512, Hh = 256;
constexpr int K1 = C1 + C2;          // 768
constexpr int ROWS = Bb * Nn;        // 65536
constexpr float EPS_W  = 1e-8f;
constexpr float EPS_BN = 1e-5f;

// ---------------------------------------------------------------------------
// Stage 1: 3-NN search + inverse-distance weights
// ---------------------------------------------------------------------------
__global__ __launch_bounds__(256)
void knn3_kernel(const float* __restrict__ unknown,  // (B,N,3)
                 const float* __restrict__ known,    // (B,M,3)
                 int* __restrict__ idx3,             // (B*N,3)
                 float* __restrict__ w3)             // (B*N,3)
{
    __shared__ float kx[Mm], ky[Mm], kz[Mm];
    const int b = blockIdx.y;
    const int n = blockIdx.x * 256 + threadIdx.x;

    for (int m = threadIdx.x; m < Mm; m += 256) {
        const float* kp = known + ((size_t)b * Mm + m) * 3;
        kx[m] = kp[0]; ky[m] = kp[1]; kz[m] = kp[2];
    }
    __syncthreads();

    const float* up = unknown + ((size_t)b * Nn + n) * 3;
    const float ux = up[0], uy = up[1], uz = up[2];

    float d0 = INFINITY, d1 = INFINITY, d2 = INFINITY;
    int   i0 = 0, i1 = 0, i2 = 0;
    for (int m = 0; m < Mm; ++m) {
        float dx = ux - kx[m], dy = uy - ky[m], dz = uz - kz[m];
        float d = dx * dx + dy * dy + dz * dz;
        if (d < d0)      { d2 = d1; i2 = i1; d1 = d0; i1 = i0; d0 = d; i0 = m; }
        else if (d < d1) { d2 = d1; i2 = i1; d1 = d;  i1 = m; }
        else if (d < d2) { d2 = d;  i2 = m; }
    }
    float w0 = 1.0f / (d0 + EPS_W);
    float w1 = 1.0f / (d1 + EPS_W);
    float w2 = 1.0f / (d2 + EPS_W);
    float inv = 1.0f / (w0 + w1 + w2);

    const size_t r = (size_t)b * Nn + n;
    idx3[r * 3 + 0] = i0; idx3[r * 3 + 1] = i1; idx3[r * 3 + 2] = i2;
    w3[r * 3 + 0] = w0 * inv; w3[r * 3 + 1] = w1 * inv; w3[r * 3 + 2] = w2 * inv;
}

// ---------------------------------------------------------------------------
// Stage 2: build A matrix: feats(r, :) = [unknow_feats^T | interpolated]
// ---------------------------------------------------------------------------
__global__ __launch_bounds__(256)
void build_feats_kernel(const float* __restrict__ uf,   // (B,C1,N)
                        const float* __restrict__ kf,   // (B,C2,M)
                        const int* __restrict__ idx3,
                        const float* __restrict__ w3,
                        float* __restrict__ feats)      // (ROWS, K1)
{
    __shared__ int   sid[3];
    __shared__ float swt[3];
    const size_t r = blockIdx.x;              // one row per block
    const int n = (int)(r & (Nn - 1));
    const int b = (int)(r >> 12);

    if (threadIdx.x < 3) {
        sid[threadIdx.x] = idx3[r * 3 + threadIdx.x];
        swt[threadIdx.x] = w3[r * 3 + threadIdx.x];
    }
    __syncthreads();

    for (int c = threadIdx.x; c < K1; c += 256) {
        float v;
        if (c < C1) {
            v = uf[((size_t)b * C1 + c) * Nn + n];
        } else {
            const int c2 = c - C1;
            const float* base = kf + ((size_t)b * C2 + c2) * Mm;
            v = swt[0] * base[sid[0]] + swt[1] * base[sid[1]] + swt[2] * base[sid[2]];
        }
        feats[r * K1 + c] = v;
    }
}

// ---------------------------------------------------------------------------
// Stage 3/5: fp32 WMMA GEMM: C(ROWS x NOUT) = A(ROWS x K) * W(NOUT x K)^T
// 8 waves/block; wave -> 16-row strip; block -> 64 output cols (4 acc tiles).
// Fragment layouts per CDNA5 ISA 7.12.2 (vgpr j <-> k = j + 2*half for A/B;
// C/D: vgpr i <-> row i + 8*half, lane%16 = col).
// ---------------------------------------------------------------------------
template <int K, int NOUT>
__global__ __launch_bounds__(256)
void gemm_wmma_kernel(const float* __restrict__ A,
                      const float* __restrict__ W,
                      float* __restrict__ C)
{
    const int lane = threadIdx.x & 31;
    const int wave = threadIdx.x >> 5;
    const int half = lane >> 4;
    const int l16  = lane & 15;

    const int row0 = (blockIdx.x * 8 + wave) * 16;
    const int col0 = blockIdx.y * 64;

    const float* a_base = A + (size_t)(row0 + l16) * K + 2 * half;
    const float* b_base0 = W + (size_t)(col0 +  0 + l16) * K + 2 * half;
    const float* b_base1 = W + (size_t)(col0 + 16 + l16) * K + 2 * half;
    const float* b_base2 = W + (size_t)(col0 + 32 + l16) * K + 2 * half;
    const float* b_base3 = W + (size_t)(col0 + 48 + l16) * K + 2 * half;

    v8f acc0 = {0.f,0.f,0.f,0.f,0.f,0.f,0.f,0.f};
    v8f acc1 = acc0, acc2 = acc0, acc3 = acc0;

    for (int k0 = 0; k0 < K; k0 += 4) {
        v2f a  = *(const v2f*)(a_base  + k0);
        v2f w0 = *(const v2f*)(b_base0 + k0);
        v2f w1 = *(const v2f*)(b_base1 + k0);
        v2f w2 = *(const v2f*)(b_base2 + k0);
        v2f w3v = *(const v2f*)(b_base3 + k0);
        acc0 = __builtin_amdgcn_wmma_f32_16x16x4_f32(false, a, false, w0,  (short)0, acc0, false, false);
        acc1 = __builtin_amdgcn_wmma_f32_16x16x4_f32(false, a, false, w1,  (short)0, acc1, false, false);
        acc2 = __builtin_amdgcn_wmma_f32_16x16x4_f32(false, a, false, w2,  (short)0, acc2, false, false);
        acc3 = __builtin_amdgcn_wmma_f32_16x16x4_f32(false, a, false, w3v, (short)0, acc3, false, false);
    }

#pragma unroll
    for (int i = 0; i < 8; ++i) {
        const size_t m = (size_t)(row0 + i + 8 * half);
        C[m * NOUT + col0 +  0 + l16] = acc0[i];
        C[m * NOUT + col0 + 16 + l16] = acc1[i];
        C[m * NOUT + col0 + 32 + l16] = acc2[i];
        C[m * NOUT + col0 + 48 + l16] = acc3[i];
    }
}

// ---------------------------------------------------------------------------
// BatchNorm: deterministic two-pass per-channel sum / sumsq reduction
// ---------------------------------------------------------------------------
constexpr int BN_BLOCKS = 256;                 // ROWS / 256 rows per block

__global__ __launch_bounds__(256)
void bn_partial_kernel(const float* __restrict__ x,   // (ROWS, 256)
                       float* __restrict__ partial)   // (BN_BLOCKS, 512)
{
    const int c  = threadIdx.x;                // channel
    const int r0 = blockIdx.x * (ROWS / BN_BLOCKS);
    float s = 0.f, ss = 0.f;
    for (int r = 0; r < ROWS / BN_BLOCKS; ++r) {
        float v = x[(size_t)(r0 + r) * Hh + c];
        s += v; ss += v * v;
    }
    partial[(size_t)blockIdx.x * 512 + c]       = s;
    partial[(size_t)blockIdx.x * 512 + 256 + c] = ss;
}

__global__ __launch_bounds__(512)
void bn_reduce_kernel(const float* __restrict__ partial,
                      float* __restrict__ stats)       // [512]: sums then sumsq
{
    const int t = threadIdx.x;
    float s = 0.f;
    for (int k = 0; k < BN_BLOCKS; ++k) s += partial[(size_t)k * 512 + t];
    stats[t] = s;
}

__global__ __launch_bounds__(256)
void bn_relu_inplace_kernel(float* __restrict__ x,
                            const float* __restrict__ stats,
                            const float* __restrict__ gamma,
                            const float* __restrict__ beta)
{
    const size_t tid = (size_t)blockIdx.x * 256 + threadIdx.x;
    const int c = (int)(tid & (Hh - 1));
    const float mean = stats[c] * (1.0f / ROWS);
    const float var  = stats[256 + c] * (1.0f / ROWS) - mean * mean;
    float v = (x[tid] - mean) * rsqrtf(var + EPS_BN) * gamma[c] + beta[c];
    x[tid] = v > 0.f ? v : 0.f;
}

__global__ __launch_bounds__(256)
void bn_relu_transpose_kernel(const float* __restrict__ x,   // (ROWS, H)
                              const float* __restrict__ stats,
                              const float* __restrict__ gamma,
                              const float* __restrict__ beta,
                              float* __restrict__ out)       // (B, H, N)
{
    const size_t tid = (size_t)blockIdx.x * 256 + threadIdx.x;
    const int c = (int)(tid & (Hh - 1));
    const size_t r = tid >> 8;
    const int n = (int)(r & (Nn - 1));
    const int b = (int)(r >> 12);
    const float mean = stats[c] * (1.0f / ROWS);
    const float var  = stats[256 + c] * (1.0f / ROWS) - mean * mean;
    float v = (x[tid] - mean) * rsqrtf(var + EPS_BN) * gamma[c] + beta[c];
    out[((size_t)b * Hh + c) * Nn + n] = v > 0.f ? v : 0.f;
}

// ---------------------------------------------------------------------------
// Launch
// ---------------------------------------------------------------------------
extern "C" void kernel_launch(void* const* d_in, const int* in_sizes, int n_in,
                              void* d_out, int out_size, void* d_ws, size_t ws_size,
                              hipStream_t stream) {
    const float* unknown      = (const float*)d_in[0];
    const float* known        = (const float*)d_in[1];
    const float* unknow_feats = (const float*)d_in[2];
    const float* known_feats  = (const float*)d_in[3];
    const float* w1 = (const float*)d_in[4];
    const float* g1 = (const float*)d_in[5];
    const float* b1 = (const float*)d_in[6];
    const float* w2 = (const float*)d_in[7];
    const float* g2 = (const float*)d_in[8];
    const float* b2 = (const float*)d_in[9];
    float* out = (float*)d_out;

    char* ws = (char*)d_ws;
    size_t off = 0;
    int*   idx3  = (int*)(ws + off);   off += (size_t)ROWS * 3 * sizeof(int);
    float* w3    = (float*)(ws + off); off += (size_t)ROWS * 3 * sizeof(float);
    float* feats = (float*)(ws + off); off += (size_t)ROWS * K1 * sizeof(float);
    float* h1    = (float*)(ws + off); off += (size_t)ROWS * Hh * sizeof(float);
    float* partial = (float*)(ws + off); off += (size_t)BN_BLOCKS * 512 * sizeof(float);
    float* stats   = (float*)(ws + off); off += 512 * sizeof(float);
    float* h2 = feats;   // feats is dead after GEMM1; reuse as GEMM2 output

    // 1. 3-NN + weights
    knn3_kernel<<<dim3(Nn / 256, Bb), 256, 0, stream>>>(unknown, known, idx3, w3);
    // 2. interpolation + concat -> A matrix
    build_feats_kernel<<<ROWS, 256, 0, stream>>>(unknow_feats, known_feats, idx3, w3, feats);
    // 3. GEMM1 (K=768) via fp32 WMMA
    gemm_wmma_kernel<K1, Hh><<<dim3(ROWS / 128, Hh / 64), 256, 0, stream>>>(feats, w1, h1);
    // 4. BN1 + ReLU (in place)
    bn_partial_kernel<<<BN_BLOCKS, 256, 0, stream>>>(h1, partial);
    bn_reduce_kernel<<<1, 512, 0, stream>>>(partial, stats);
    bn_relu_inplace_kernel<<<(ROWS * Hh) / 256, 256, 0, stream>>>(h1, stats, g1, b1);
    // 5. GEMM2 (K=256) via fp32 WMMA
    gemm_wmma_kernel<Hh, Hh><<<dim3(ROWS / 128, Hh / 64), 256, 0, stream>>>(h1, w2, h2);
    // 6. BN2 + ReLU + transpose to (B,H,N)
    bn_partial_kernel<<<BN_BLOCKS, 256, 0, stream>>>(h2, partial);
    bn_reduce_kernel<<<1, 512, 0, stream>>>(partial, stats);
    bn_relu_transpose_kernel<<<(ROWS * Hh) / 256, 256, 0, stream>>>(h2, stats, g2, b2, out);
}